// SkipRouter_45346264711638
// MI455X (gfx1250) — compile-verified
//
#include <hip/hip_runtime.h>
#include <math.h>

// ---------------------------------------------------------------------------
// SkipRouter for MI455X (gfx1250, wave32).
//   logits = x[16384,2048] @ W[64,2048]^T       (fp32 WMMA 16x16x4)
//   softmax over 64 experts, top-2, skip @ 0.1, renormalize
// Memory-bound: 134 MB of x @ 23.3 TB/s ~= 5.8 us floor. fp32 WMMA keeps
// exact routing decisions while easily pacing the HBM stream.
// ---------------------------------------------------------------------------

typedef float v2f __attribute__((ext_vector_type(2)));
typedef float v8f __attribute__((ext_vector_type(8)));

#define T_TOK 16384
#define H_DIM 2048
#define E_EXP 64
#define WAVES_PER_BLOCK 8
#define TOK_PER_WAVE 32          // two 16-row M-tiles per wave
#define TOK_PER_BLOCK (WAVES_PER_BLOCK * TOK_PER_WAVE)   // 256
#define LDS_STRIDE 67            // 3t+e bank pattern: conflict-free both phases

__device__ __forceinline__ v2f load_frag(const float* __restrict__ base, int k0) {
  // base already points at row*H + koff (koff = (lane>>4)*2)
  return *(const v2f*)(base + k0);
}

__global__ __launch_bounds__(TOK_PER_BLOCK)
void skip_router_kernel(const float* __restrict__ x,
                        const float* __restrict__ w,
                        float* __restrict__ out) {
  __shared__ float lds[TOK_PER_BLOCK * LDS_STRIDE];

  const int lane = threadIdx.x & 31;
  const int wave = threadIdx.x >> 5;
  const int lo   = lane & 15;      // row (A) / col (B) within tile
  const int hi   = lane >> 4;      // selects K pair {0,1} vs {2,3}
  const int koff = hi << 1;

  const int m_base = blockIdx.x * TOK_PER_BLOCK + wave * TOK_PER_WAVE;

  // A fragments: rows m_base+lo and m_base+16+lo of x
  const float* xr0 = x + (size_t)(m_base + lo)      * H_DIM + koff;
  const float* xr1 = x + (size_t)(m_base + 16 + lo) * H_DIM + koff;
  // B fragments: expert rows (nt*16 + lo) of W — identical layout helper
  const float* wr0 = w + (size_t)( 0 + lo) * H_DIM + koff;
  const float* wr1 = w + (size_t)(16 + lo) * H_DIM + koff;
  const float* wr2 = w + (size_t)(32 + lo) * H_DIM + koff;
  const float* wr3 = w + (size_t)(48 + lo) * H_DIM + koff;

  v8f acc[2][4];
#pragma unroll
  for (int mt = 0; mt < 2; ++mt)
#pragma unroll
    for (int nt = 0; nt < 4; ++nt)
      acc[mt][nt] = (v8f){0.f, 0.f, 0.f, 0.f, 0.f, 0.f, 0.f, 0.f};

#pragma unroll 4
  for (int k0 = 0; k0 < H_DIM; k0 += 4) {
    v2f a0 = load_frag(xr0, k0);
    v2f a1 = load_frag(xr1, k0);
    v2f b0 = load_frag(wr0, k0);
    v2f b1 = load_frag(wr1, k0);
    v2f b2 = load_frag(wr2, k0);
    v2f b3 = load_frag(wr3, k0);

    acc[0][0] = __builtin_amdgcn_wmma_f32_16x16x4_f32(false, a0, false, b0, (short)0, acc[0][0], false, false);
    acc[1][0] = __builtin_amdgcn_wmma_f32_16x16x4_f32(false, a1, false, b0, (short)0, acc[1][0], false, false);
    acc[0][1] = __builtin_amdgcn_wmma_f32_16x16x4_f32(false, a0, false, b1, (short)0, acc[0][1], false, false);
    acc[1][1] = __builtin_amdgcn_wmma_f32_16x16x4_f32(false, a1, false, b1, (short)0, acc[1][1], false, false);
    acc[0][2] = __builtin_amdgcn_wmma_f32_16x16x4_f32(false, a0, false, b2, (short)0, acc[0][2], false, false);
    acc[1][2] = __builtin_amdgcn_wmma_f32_16x16x4_f32(false, a1, false, b2, (short)0, acc[1][2], false, false);
    acc[0][3] = __builtin_amdgcn_wmma_f32_16x16x4_f32(false, a0, false, b3, (short)0, acc[0][3], false, false);
    acc[1][3] = __builtin_amdgcn_wmma_f32_16x16x4_f32(false, a1, false, b3, (short)0, acc[1][3], false, false);
  }

  // --- Spill logits to LDS following the 16x16 f32 C/D layout --------------
  // VGPR v of acc[mt][nt] on lane L holds  M = mt*16 + v + 8*(L>>4),
  //                                        N = nt*16 + (L&15).
#pragma unroll
  for (int mt = 0; mt < 2; ++mt) {
#pragma unroll
    for (int nt = 0; nt < 4; ++nt) {
#pragma unroll
      for (int v = 0; v < 8; ++v) {
        int tl = wave * TOK_PER_WAVE + mt * 16 + v + 8 * hi;  // token in block
        int e  = nt * 16 + lo;                                 // expert
        lds[tl * LDS_STRIDE + e] = acc[mt][nt][v];
      }
    }
  }
  __syncthreads();

  // --- Phase 2: one thread per token — softmax / top-2 / skip / renorm -----
  const int   tl  = threadIdx.x;
  const float* row = lds + tl * LDS_STRIDE;

  float best = -INFINITY, second = -INFINITY;
  int   bi = 0, si = 0;
#pragma unroll
  for (int e = 0; e < E_EXP; ++e) {
    float l = row[e];
    if (l > best)        { second = best; si = bi; best = l; bi = e; }
    else if (l > second) { second = l; si = e; }
  }

  float z = 0.f;
#pragma unroll
  for (int e = 0; e < E_EXP; ++e)
    z += __expf(row[e] - best);        // best == row max

  float inv_z = 1.0f / z;
  float p1 = inv_z;                    // exp(best-best)/z
  float p2 = __expf(second - best) * inv_z;

  float g1 = (p1 >= 0.1f) ? p1 : 0.f;
  float g2 = (p2 >= 0.1f) ? p2 : 0.f;
  float den = g1 + g2;
  float r  = (den > 0.f) ? (1.0f / den) : 1.0f;
  g1 *= r;
  g2 *= r;

  const int token = blockIdx.x * TOK_PER_BLOCK + tl;
  out[(size_t)token * 2 + 0] = g1;
  out[(size_t)token * 2 + 1] = g2;
  float* oidx = out + (size_t)T_TOK * 2;
  oidx[(size_t)token * 2 + 0] = (float)bi;
  oidx[(size_t)token * 2 + 1] = (float)si;
}

extern "C" void kernel_launch(void* const* d_in, const int* in_sizes, int n_in,
                              void* d_out, int out_size, void* d_ws, size_t ws_size,
                              hipStream_t stream) {
  (void)in_sizes; (void)n_in; (void)out_size; (void)d_ws; (void)ws_size;
  const float* x = (const float*)d_in[0];
  const float* w = (const float*)d_in[1];
  float* out = (float*)d_out;

  dim3 grid(T_TOK / TOK_PER_BLOCK);   // 64 blocks
  dim3 block(TOK_PER_BLOCK);          // 256 threads = 8 waves (wave32)
  skip_router_kernel<<<grid, block, 0, stream>>>(x, w, out);
}